// CANN_81252191306279
// MI455X (gfx1250) — compile-verified
//
#include <hip/hip_runtime.h>
#include <cstdint>

#define THREADS 256
#define MAX_BLOCKS 4096

typedef float v2f __attribute__((ext_vector_type(2)));

__device__ __forceinline__ v2f splat2(float x) { v2f v = {x, x}; return v; }
__device__ __forceinline__ int imin(int a, int b) { return a < b ? a : b; }

__device__ __forceinline__ v2f rcp_v2(v2f x) {
  v2f r;
  r.x = __builtin_amdgcn_rcpf(x.x);      // v_rcp_f32 (trans unit, ~1 ulp)
  r.y = __builtin_amdgcn_rcpf(x.y);
  return r;
}

__device__ __forceinline__ v2f exp2_v2(v2f x) {
  v2f r;
  r.x = __builtin_amdgcn_exp2f(x.x);     // v_exp_f32 (trans unit)
  r.y = __builtin_amdgcn_exp2f(x.y);
  return r;
}

// Truncating a generic (flat) pointer to a __shared__ object yields the
// wave-relative LDS byte offset on gfx1250 (flat-LDS aperture: the low 32 bits are the LDS address).
__device__ __forceinline__ uint32_t lds_addr32(const void* p) {
  return (uint32_t)(uintptr_t)p;
}

// gfx1250 async data mover: per-lane 16B copy global -> LDS, tracked by ASYNCcnt.
// GVS addressing: mem = SGPR64 base + VGPR32 byte offset.
__device__ __forceinline__ void async_copy_b128(uint32_t lds_byte,
                                                uint32_t gl_byte_off,
                                                const float* base) {
  asm volatile("global_load_async_to_lds_b128 %0, %1, %2"
               :
               : "v"(lds_byte), "v"(gl_byte_off), "s"(base)
               : "memory");
}

// 12 folded coefficients. The global 2.0x of P1 is folded into the linear
// terms (c0,c2,a0,a2 / d0,d2,e0,e2); log2(e) is folded into the exp rates
// (b0,b2 / f0,f2) so we can use native exp2 directly.
struct Coefs {
  float c0, c2, a0, b0, a2, b2;   // 2*dPsi/dI1 terms
  float d0, d2, e0, f0, e2, f2;   // 2*dPsi/dI2 terms
};

// Two samples at a time -> whole chain lowers to v_pk_{fma,mul,add}_f32.
__device__ __forceinline__ v2f p1_pair(v2f s, const Coefs& k) {
  v2f inv_s  = rcp_v2(s);
  v2f s2     = s * s;
  v2f inv_s2 = inv_s * inv_s;
  v2f r1 = __builtin_elementwise_fma(splat2(2.0f), inv_s, s2) + splat2(-3.0f); // I1-3
  v2f r2 = __builtin_elementwise_fma(splat2(2.0f), s, inv_s2) + splat2(-3.0f); // I2-3

  v2f e1 = exp2_v2(r1 * splat2(k.b0));
  v2f e2 = exp2_v2((r1 * r1) * splat2(k.b2));
  v2f g1 = __builtin_elementwise_fma(splat2(k.c2), r1, splat2(k.c0));
  g1 = __builtin_elementwise_fma(splat2(k.a0), e1, g1);
  g1 = __builtin_elementwise_fma(splat2(k.a2) * r1, e2, g1);

  v2f e3 = exp2_v2(r2 * splat2(k.f0));
  v2f e4 = exp2_v2((r2 * r2) * splat2(k.f2));
  v2f g2 = __builtin_elementwise_fma(splat2(k.d2), r2, splat2(k.d0));
  g2 = __builtin_elementwise_fma(splat2(k.e0), e3, g2);
  g2 = __builtin_elementwise_fma(splat2(k.e2) * r2, e4, g2);

  // P1 = (2 dPsi/dI1 + 2 dPsi/dI2 / s) * (s - 1/s^2)   [the 2x is in g1,g2]
  return __builtin_elementwise_fma(g2, inv_s, g1) * (s - inv_s2);
}

__device__ __forceinline__ float4 p1_quad(float4 v, const Coefs& k) {
  v2f lo = {v.x, v.y};
  v2f hi = {v.z, v.w};
  v2f plo = p1_pair(lo, k);
  v2f phi = p1_pair(hi, k);
  float4 r;
  r.x = plo.x; r.y = plo.y; r.z = phi.x; r.w = phi.y;
  return r;
}

__device__ __forceinline__ float p1_scalar(float s, const Coefs& k) {
  v2f v = {s, s};
  v2f p = p1_pair(v, k);
  return p.x;
}

__device__ __forceinline__ Coefs make_coefs(const float* w_id, const float* w_e,
                                            const float* w_p) {
  const float LOG2E = 1.4426950408889634f;
  Coefs k;
  k.c0 = 2.0f * w_p[0] * w_id[0];
  k.c2 = 4.0f * w_p[2] * w_id[2];            // 2 (P1) * 2 (d r^2/dr)
  k.a0 = 2.0f * w_p[4] * w_e[0];   k.b0 = w_e[0] * LOG2E;
  k.a2 = 4.0f * w_p[6] * w_e[2];   k.b2 = w_e[2] * LOG2E;
  k.d0 = 2.0f * w_p[1] * w_id[1];
  k.d2 = 4.0f * w_p[3] * w_id[3];
  k.e0 = 2.0f * w_p[5] * w_e[1];   k.f0 = w_e[1] * LOG2E;
  k.e2 = 4.0f * w_p[7] * w_e[3];   k.f2 = w_e[3] * LOG2E;
  return k;
}

__global__ __launch_bounds__(THREADS)
void CANN_81252191306279_kernel(const float* __restrict__ stretch,
                                const float* __restrict__ w_id,   // (4)
                                const float* __restrict__ w_e,    // (4)
                                const float* __restrict__ w_p,    // (8)
                                float* __restrict__ out,
                                int n4, int n) {
  const Coefs k = make_coefs(w_id, w_e, w_p);

  // Lane-private double buffer, 2 float4 slots per stage: 16 KB / block.
  __shared__ float4 buf[2][2][THREADS];
  const int tid  = threadIdx.x;
  const int step = gridDim.x * (2 * THREADS);  // grid-stride in float4 units
  int i = blockIdx.x * (2 * THREADS) + tid;    // slot A; slot B = i + THREADS

  uint32_t lds[2][2];
  lds[0][0] = lds_addr32(&buf[0][0][tid]);
  lds[0][1] = lds_addr32(&buf[0][1][tid]);
  lds[1][0] = lds_addr32(&buf[1][0][tid]);
  lds[1][1] = lds_addr32(&buf[1][1][tid]);

  float4* __restrict__ out4 = (float4*)out;

  if (i < n4) {
    const int nm1 = n4 - 1;
    // Prologue: stage both tiles for iteration 0 (clamped -> always in range).
    async_copy_b128(lds[0][0], (uint32_t)imin(i, nm1) * 16u, stretch);
    async_copy_b128(lds[0][1], (uint32_t)imin(i + THREADS, nm1) * 16u, stretch);
    int cur = 0;
    for (; i < n4; i += step) {
      // Stage both tiles for the NEXT iteration (clamped so every pass issues
      // exactly two async ops per wave -> branch-free ASYNCcnt bookkeeping).
      int inext = i + step;
      async_copy_b128(lds[cur ^ 1][0], (uint32_t)imin(inext, nm1) * 16u, stretch);
      async_copy_b128(lds[cur ^ 1][1], (uint32_t)imin(inext + THREADS, nm1) * 16u,
                      stretch);
      // Allow the two just-issued copies to stay in flight; older ones are done.
      asm volatile("s_wait_asynccnt 0x2" ::: "memory");

      float4 va = buf[cur][0][tid];            // ds_load_b128
      float4 vb = buf[cur][1][tid];            // ds_load_b128
      float4 ra = p1_quad(va, k);
      float4 rb = p1_quad(vb, k);
      out4[i] = ra;                            // i < n4 guaranteed by loop cond
      int i2 = i + THREADS;
      if (i2 < n4) out4[i2] = rb;
      cur ^= 1;
    }
    asm volatile("s_wait_asynccnt 0x0" ::: "memory");  // drain before reusing LDS
  }

  // Scalar tail for n % 4 elements (empty for N = 16M, kept for generality).
  const int gstride = gridDim.x * THREADS;
  for (int j = n4 * 4 + blockIdx.x * THREADS + tid; j < n; j += gstride) {
    out[j] = p1_scalar(stretch[j], k);
  }
}

extern "C" void kernel_launch(void* const* d_in, const int* in_sizes, int n_in,
                              void* d_out, int out_size, void* d_ws, size_t ws_size,
                              hipStream_t stream) {
  (void)n_in; (void)out_size; (void)d_ws; (void)ws_size;
  const float* stretch = (const float*)d_in[0];
  const float* w_id    = (const float*)d_in[1];
  const float* w_e     = (const float*)d_in[2];
  const float* w_p     = (const float*)d_in[3];
  float* out = (float*)d_out;

  int n  = in_sizes[0];
  int n4 = n >> 2;
  int work = (n4 > 0) ? n4 : 1;
  int blocks = (work + 2 * THREADS - 1) / (2 * THREADS);
  if (blocks > MAX_BLOCKS) blocks = MAX_BLOCKS;
  if (blocks < 1) blocks = 1;

  CANN_81252191306279_kernel<<<blocks, THREADS, 0, stream>>>(
      stretch, w_id, w_e, w_p, out, n4, n);
}